// SubSampling_20839181320837
// MI455X (gfx1250) — compile-verified
//
#include <hip/hip_runtime.h>

// ---------------------------------------------------------------------------
// 2x2 sum-pool + affine, pure HBM-streaming kernel for MI455X (gfx1250).
// x: (1024 planes, 512, 512) f32 -> out: (1024, 256, 256) f32
// Roofline: 1.25 GiB traffic / 23.3 TB/s ~ 55us; compute is negligible.
// Data path: CDNA5 async global->LDS copies (ASYNCcnt) + NT 64b stores.
// ---------------------------------------------------------------------------

#define AS1 __attribute__((address_space(1)))
#define AS3 __attribute__((address_space(3)))

#if defined(__AMDGCN__) && \
    __has_builtin(__builtin_amdgcn_global_load_async_to_lds_b128) && \
    __has_builtin(__builtin_amdgcn_s_wait_asynccnt)
#define USE_ASYNC_LDS 1
#else
#define USE_ASYNC_LDS 0
#endif

typedef float v2f __attribute__((ext_vector_type(2)));
typedef float v4f __attribute__((ext_vector_type(4)));
typedef int   b128_t __attribute__((vector_size(16)));   // builtin's pointee type

namespace {
constexpr int TPB             = 256;   // 8 wave32s
constexpr int W               = 512;   // input row length
constexpr int OW              = 256;   // output row length
constexpr int OROWS_PER_BLOCK = 8;     // output rows per block
constexpr int IN_FLOATS_BLK   = 2 * OROWS_PER_BLOCK * W;   // 8192 (32 KB)
constexpr int OUT_FLOATS_BLK  = OROWS_PER_BLOCK * OW;      // 2048 (8 KB)
} // namespace

__global__ __launch_bounds__(TPB) void
SubSampling_pool_affine(const float* __restrict__ x,
                        const float* __restrict__ coefficient,
                        const float* __restrict__ bias,
                        float* __restrict__ out) {
    const int tid = threadIdx.x;
    const float cf = coefficient[0];
    const float bs = bias[0];

#if USE_ASYNC_LDS
    // ---- Stage one contiguous 32 KB input chunk into LDS via async DMA ----
    __shared__ float tile[IN_FLOATS_BLK];  // 16 input rows x 512 floats

    const size_t in_base = (size_t)blockIdx.x * IN_FLOATS_BLK;
    const AS1 char* gbase = (const AS1 char*)(x + in_base);  // generic -> AS1
    AS3 char*       lbase = (AS3 char*)tile;                 // generic -> AS3

#pragma unroll
    for (int k = 0; k < 8; ++k) {
        const int o16 = k * TPB + tid;  // 16-byte unit index, wave-contiguous
        __builtin_amdgcn_global_load_async_to_lds_b128(
            (AS1 b128_t*)(gbase + (size_t)o16 * 16),
            (AS3 b128_t*)(lbase + o16 * 16),
            /*offset=*/0, /*cpol=*/0);
    }
    __builtin_amdgcn_s_wait_asynccnt(0);
    __syncthreads();  // LDS regions are written by all 8 waves

    // ---- Pool from LDS, affine, NT store ----
    const size_t out_base = (size_t)blockIdx.x * OUT_FLOATS_BLK;
#pragma unroll
    for (int k = 0; k < 4; ++k) {
        const int q   = k * TPB + tid;  // float2 id within block, 0..1023
        const int c2  = q & 127;        // float2 column in output row
        const int r   = q >> 7;         // local output row, 0..7
        // input row pair 2r, 2r+1 lives at tile[r*1024 .. r*1024+1023]
        const v4f a = *(const v4f*)&tile[r * 1024 + c2 * 4];
        const v4f b = *(const v4f*)&tile[r * 1024 + 512 + c2 * 4];
        v2f o;
        o.x = cf * ((a.x + a.y) + (b.x + b.y)) + bs;
        o.y = cf * ((a.z + a.w) + (b.z + b.w)) + bs;
        __builtin_nontemporal_store(o, (v2f*)(out + out_base) + q);
    }
#else
    // ---- Fallback: direct NT 128-bit loads, no LDS staging ----
#pragma unroll
    for (int k = 0; k < 4; ++k) {
        const size_t G  = (size_t)blockIdx.x * 1024 + (size_t)(k * TPB + tid);
        const size_t in0 = ((G >> 7) << 10) + ((G & 127) << 2);
        const v4f a = __builtin_nontemporal_load((const v4f*)(x + in0));
        const v4f b = __builtin_nontemporal_load((const v4f*)(x + in0 + W));
        v2f o;
        o.x = cf * ((a.x + a.y) + (b.x + b.y)) + bs;
        o.y = cf * ((a.z + a.w) + (b.z + b.w)) + bs;
        __builtin_nontemporal_store(o, (v2f*)out + G);
    }
#endif
}

extern "C" void kernel_launch(void* const* d_in, const int* in_sizes, int n_in,
                              void* d_out, int out_size, void* d_ws, size_t ws_size,
                              hipStream_t stream) {
    (void)n_in; (void)d_ws; (void)ws_size; (void)in_sizes;
    const float* x    = (const float*)d_in[0];
    const float* coef = (const float*)d_in[1];
    const float* bias = (const float*)d_in[2];
    float* out        = (float*)d_out;

    const int blocks = out_size / OUT_FLOATS_BLK;  // 67,108,864 / 2048 = 32,768
    SubSampling_pool_affine<<<blocks, TPB, 0, stream>>>(x, coef, bias, out);
}